// CenterLoss_15015205667266
// MI455X (gfx1250) — compile-verified
//
#include <hip/hip_runtime.h>

// CenterLoss on MI455X (gfx1250):
//   dist[b] = clip(||x_b - centers[labels[b]]||^2, 1e-12, 1e12); out = mean(dist)
// Memory-bound (~4 MB @ 23.3 TB/s); the K-dim reduction is done on the matrix
// pipe with V_WMMA_F32_16X16X4_F32 (exact f32 math): one wave owns a 16-row
// tile, computes diag(E * E^T) with E = x_tile - c_tile accumulated over K=256.

typedef float v2f __attribute__((ext_vector_type(2)));
typedef float v8f __attribute__((ext_vector_type(8)));

#define FEAT_DIM 256
#define BATCH    2048
#define ROWS_PER_WAVE 16
#define NUM_TILES (BATCH / ROWS_PER_WAVE)   // 128 waves total

__global__ __launch_bounds__(32) void center_loss_partial(
    const float* __restrict__ x,        // [BATCH, FEAT_DIM]
    const int*   __restrict__ labels,   // [BATCH]
    const float* __restrict__ centers,  // [NUM_CLASSES, FEAT_DIM]
    float*       __restrict__ partial)  // [NUM_TILES]
{
    const int lane = threadIdx.x;        // 0..31 (wave32)
    const int tile = blockIdx.x;         // 0..127
    const int m    = lane & 15;          // matrix row (M or N) owned by this lane
    const int h    = lane >> 4;          // half: 0 -> K pair {0,1}, 1 -> {2,3}
    const int row  = tile * ROWS_PER_WAVE + m;
    const int lab  = labels[row];

    // A 16x4 f32 layout: lanes 0-15 hold (M=lane, K=0..1), lanes 16-31 (M=lane-16, K=2..3).
    // B 4x16 f32 is the mirror (lane indexes N), so the same float2 of E feeds both.
    const float* xr = x       + (size_t)row * FEAT_DIM + 2 * h;
    const float* cr = centers + (size_t)lab * FEAT_DIM + 2 * h;

    v8f acc = {};   // 16x16 f32 accumulator; only the diagonal is used
    #pragma unroll 4
    for (int k = 0; k < FEAT_DIM; k += 4) {
        v2f xv = *(const v2f*)(xr + k);
        v2f cv = *(const v2f*)(cr + k);
        v2f e  = xv - cv;
        // D = E(16x4) * E(4x16) + C  -> accumulates e_m . e_n over K
        acc = __builtin_amdgcn_wmma_f32_16x16x4_f32(
                  /*neg_a=*/false, e, /*neg_b=*/false, e,
                  /*c_mod=*/(short)0, acc, /*reuse_a=*/false, /*reuse_b=*/false);
    }

    // Diagonal extraction per C/D layout:
    //   VGPR j, lanes 0-15: (M=j,   N=lane)    -> diag m=0..7  at lane=m,    elem=m
    //   VGPR j, lanes 16-31:(M=8+j, N=lane-16) -> diag m=8..15 at lane=m+16, elem=m-8
    int idx = -1;
    if (lane < 8)        idx = lane;        // m = lane
    else if (lane >= 24) idx = lane - 24;   // m = lane - 16
    float val = 0.f;
    #pragma unroll
    for (int j = 0; j < 8; ++j)
        if (idx == j) val = acc[j];
    if (idx >= 0)
        val = fminf(fmaxf(val, 1e-12f), 1e12f);   // reference clip
    // lanes 8..23 contribute 0

    // Deterministic wave32 tree reduction of the 16 clipped distances
    #pragma unroll
    for (int off = 16; off > 0; off >>= 1)
        val += __shfl_xor(val, off, 32);

    if (lane == 0) partial[tile] = val;
}

__global__ __launch_bounds__(32) void center_loss_reduce(
    const float* __restrict__ partial,  // [NUM_TILES]
    float*       __restrict__ out)      // [1]
{
    const int lane = threadIdx.x;
    float s = 0.f;
    #pragma unroll
    for (int i = 0; i < NUM_TILES; i += 32)   // fixed order -> deterministic
        s += partial[i + lane];
    #pragma unroll
    for (int off = 16; off > 0; off >>= 1)
        s += __shfl_xor(s, off, 32);
    if (lane == 0) out[0] = s * (1.0f / (float)BATCH);
}

extern "C" void kernel_launch(void* const* d_in, const int* in_sizes, int n_in,
                              void* d_out, int out_size, void* d_ws, size_t ws_size,
                              hipStream_t stream) {
    const float* x       = (const float*)d_in[0];
    const int*   labels  = (const int*)  d_in[1];   // harness: integer -> const int*
    const float* centers = (const float*)d_in[2];
    float*       out     = (float*)d_out;
    float*       partial = (float*)d_ws;            // 128 floats of scratch

    center_loss_partial<<<NUM_TILES, 32, 0, stream>>>(x, labels, centers, partial);
    center_loss_reduce<<<1, 32, 0, stream>>>(partial, out);
}